// DiversityLoss_206158430576
// MI455X (gfx1250) — compile-verified
//
#include <hip/hip_runtime.h>

typedef __attribute__((ext_vector_type(2))) float v2f;
typedef __attribute__((ext_vector_type(4))) float v4f;
typedef __attribute__((ext_vector_type(8))) float v8f;

#define N_PTS 8192
#define DIMS 64
#define TB 256                          // N_PTS / 32 : 32x32 blocks
#define BLOCK_PAIRS 32896               // TB*(TB+1)/2
#define WAVES_PER_BLOCK 8
#define BLOCKS_GRAM (BLOCK_PAIRS / WAVES_PER_BLOCK)  // 4112 exactly
#define NEG_SCALE_LOG2E (-0.14426950408889634f)      // -0.1 * log2(e)
#define INV_NPAIRS (1.0f / 33550336.0f)              // 1 / (8192*8191/2)

// ---------------------------------------------------------------------------
// Kernel 1: row squared-norms (8192 rows x 64 floats; fully L2 resident)
// ---------------------------------------------------------------------------
__global__ __launch_bounds__(256) void norms_kernel(const float* __restrict__ x,
                                                    float* __restrict__ norms) {
    int i = blockIdx.x * 256 + threadIdx.x;
    if (i < N_PTS) {
        const v4f* row = (const v4f*)(x + (size_t)i * DIMS);
        float s = 0.0f;
#pragma unroll
        for (int k = 0; k < DIMS / 4; ++k) {
            v4f v = row[k];
            s += v.x * v.x + v.y * v.y + v.z * v.z + v.w * v.w;
        }
        norms[i] = s;
    }
}

// ---------------------------------------------------------------------------
// Kernel 2: one wave32 per 32x32 block-pair of the upper triangle.
// 2x2 register-blocked Gram: 64 chained V_WMMA_F32_16X16X4_F32 from
// 2 A-tiles + 2 B-tiles (8 FLOP/B from L1), fused branchless sqrt/exp
// epilogue, deterministic block partial sums.
// ---------------------------------------------------------------------------
__device__ __forceinline__ int row_start(int t) {
    // first linear pair index of block-row t (upper triangle incl. diagonal)
    return t * TB - (t * (t - 1)) / 2;
}

__device__ __forceinline__ float tile_term_sum(const v8f& c, const float* ni,
                                               float nj, int ibase, int j) {
    float s = 0.0f;
#pragma unroll
    for (int r = 0; r < 8; ++r) {
        float sq = ni[r] + nj - 2.0f * c[r];
        float d  = __builtin_sqrtf(fmaxf(sq, 0.0f));
        float t  = exp2f(NEG_SCALE_LOG2E * d);   // v_mul + v_exp_f32
        s += (j > ibase + r) ? t : 0.0f;         // branchless v_cndmask
    }
    return s;
}

__global__ __launch_bounds__(256) void gram_exp_kernel(const float* __restrict__ x,
                                                       const float* __restrict__ norms,
                                                       float* __restrict__ partial) {
    __shared__ float red[WAVES_PER_BLOCK];

    const int lane = threadIdx.x & 31;
    const int wave = threadIdx.x >> 5;
    const int p    = blockIdx.x * WAVES_PER_BLOCK + wave;   // block-pair index

    // ---- map p -> (tbi, tbj) with tbj >= tbi (fp32 sqrt + exact fixup)
    float disc = (float)((2 * TB + 1) * (2 * TB + 1)) - 8.0f * (float)p;
    int ti = (int)(((float)(2 * TB + 1) - sqrtf(disc)) * 0.5f);
    ti = min(max(ti, 0), TB - 1);
    while (ti + 1 < TB && row_start(ti + 1) <= p) ++ti;
    while (ti > 0 && row_start(ti) > p) --ti;
    const int tj = ti + (p - row_start(ti));

    const int i0 = ti * 32;
    const int j0 = tj * 32;

    // ---- WMMA fp32 16x16x4 operand layout (ISA 7.12.2):
    // A: lanes 0-15 hold K={0,1}, lanes 16-31 hold K={2,3}; M = lane%16.
    // B mirrors with N = lane%16. Both A and B tiles are rows of x.
    const int mn  = lane & 15;
    const int khi = lane >> 4;
    const float* arow0 = x + (size_t)(i0 + mn) * DIMS + khi * 2;
    const float* arow1 = arow0 + 16 * DIMS;
    const float* brow0 = x + (size_t)(j0 + mn) * DIMS + khi * 2;
    const float* brow1 = brow0 + 16 * DIMS;

    // Hoisted epilogue operands: 2 scalar norms[j] + 2 runs of 8 consecutive
    // norms[i] as four aligned b128 loads.
    const int   ja  = j0 + mn;
    const int   jb  = ja + 16;
    const float nja = norms[ja];
    const float njb = norms[jb];
    const v4f*  nv  = (const v4f*)(norms + i0 + khi * 8);   // 32B aligned
    const v4f   n0 = nv[0], n1 = nv[1];                     // rows i0..i0+15
    const v4f   n2 = nv[4], n3 = nv[5];                     // rows i0+16..i0+31
    const float nia[8] = {n0.x, n0.y, n0.z, n0.w, n1.x, n1.y, n1.z, n1.w};
    const float nib[8] = {n2.x, n2.y, n2.z, n2.w, n3.x, n3.y, n3.z, n3.w};

    // ---- 2x2 register-blocked WMMA: 4 independent accumulators break the
    // serial accumulation chain (each acc's next WMMA is 3 insts away).
    v8f c00 = {}, c01 = {}, c10 = {}, c11 = {};
#pragma unroll
    for (int kk = 0; kk < 16; ++kk) {
        v2f a0 = *(const v2f*)(arow0 + kk * 4);
        v2f a1 = *(const v2f*)(arow1 + kk * 4);
        v2f b0 = *(const v2f*)(brow0 + kk * 4);
        v2f b1 = *(const v2f*)(brow1 + kk * 4);
        c00 = __builtin_amdgcn_wmma_f32_16x16x4_f32(false, a0, false, b0, (short)0, c00, false, false);
        c01 = __builtin_amdgcn_wmma_f32_16x16x4_f32(false, a0, false, b1, (short)0, c01, false, false);
        c10 = __builtin_amdgcn_wmma_f32_16x16x4_f32(false, a1, false, b0, (short)0, c10, false, false);
        c11 = __builtin_amdgcn_wmma_f32_16x16x4_f32(false, a1, false, b1, (short)0, c11, false, false);
    }

    // ---- fused branchless epilogue: C VGPR r of tile (a,b) holds
    // gram[i0+16a+r+8*khi][j0+16b+mn]. Strict upper triangle via mask
    // (also zeroes the redundant lower tile of diagonal blocks).
    const int ia = i0 + khi * 8;
    const int ib = ia + 16;
    float sum = 0.0f;
    sum += tile_term_sum(c00, nia, nja, ia, ja);
    sum += tile_term_sum(c01, nia, njb, ia, jb);
    sum += tile_term_sum(c10, nib, nja, ib, ja);
    sum += tile_term_sum(c11, nib, njb, ib, jb);

    // ---- wave32 reduction, then block reduction (fixed order => deterministic)
#pragma unroll
    for (int off = 16; off > 0; off >>= 1)
        sum += __shfl_down(sum, off, 32);
    if (lane == 0) red[wave] = sum;
    __syncthreads();
    if (threadIdx.x == 0) {
        float b = 0.0f;
#pragma unroll
        for (int w = 0; w < WAVES_PER_BLOCK; ++w) b += red[w];
        partial[blockIdx.x] = b;
    }
}

// ---------------------------------------------------------------------------
// Kernel 3: deterministic final reduction of 4112 block partials
// ---------------------------------------------------------------------------
__global__ __launch_bounds__(256) void reduce_kernel(const float* __restrict__ partial,
                                                     float* __restrict__ out) {
    __shared__ float red[256];
    float s = 0.0f;
    for (int i = threadIdx.x; i < BLOCKS_GRAM; i += 256)
        s += partial[i];
    red[threadIdx.x] = s;
    __syncthreads();
#pragma unroll
    for (int stride = 128; stride > 0; stride >>= 1) {
        if (threadIdx.x < stride) red[threadIdx.x] += red[threadIdx.x + stride];
        __syncthreads();
    }
    if (threadIdx.x == 0) out[0] = red[0] * INV_NPAIRS;
}

// ---------------------------------------------------------------------------
extern "C" void kernel_launch(void* const* d_in, const int* in_sizes, int n_in,
                              void* d_out, int out_size, void* d_ws, size_t ws_size,
                              hipStream_t stream) {
    (void)in_sizes; (void)n_in; (void)out_size; (void)ws_size;
    const float* x = (const float*)d_in[0];

    float* norms   = (float*)d_ws;          // 8192 floats
    float* partial = norms + N_PTS;         // 4112 floats (total ~49 KB of ws)

    norms_kernel<<<N_PTS / 256, 256, 0, stream>>>(x, norms);
    gram_exp_kernel<<<BLOCKS_GRAM, 256, 0, stream>>>(x, norms, partial);
    reduce_kernel<<<1, 256, 0, stream>>>(partial, (float*)d_out);
}